// GCN_43662637531292
// MI455X (gfx1250) — compile-verified
//
#include <hip/hip_runtime.h>
#include <hip/hip_bf16.h>

typedef float v2f __attribute__((ext_vector_type(2)));
typedef float v8f __attribute__((ext_vector_type(8)));

#define NEG_SLOPE 0.2f

// Monotonic float <-> int encoding so we can use integer atomicMax for float max.
__device__ __forceinline__ int enc_f(float f) {
    int i = __float_as_int(f);
    return i >= 0 ? i : (i ^ 0x7FFFFFFF);
}
__device__ __forceinline__ float dec_f(int i) {
    return __int_as_float(i >= 0 ? i : (i ^ 0x7FFFFFFF));
}

// ---------------------------------------------------------------------------
// Dense GEMM  H[n, FO] = X[n, FI] @ W[FI, FO] (+ bias), f32 WMMA 16x16x4.
// One wave32 computes a 16-row tile; FO/16 accumulators per wave.
// A layout (16x4 f32): lane = M (lane&15), VGPR v holds K = (lane>>4)*2 + v.
// B layout (4x16 f32): lane = N (lane&15), VGPR v holds K = (lane>>4)*2 + v.
// C/D layout (16x16 f32): VGPR r -> row r (+8 for lanes 16..31), col = lane&15.
// ---------------------------------------------------------------------------
template<int FI, int FO, bool BIAS>
__global__ __launch_bounds__(256) void wmma_gemm(
    const float* __restrict__ X, const float* __restrict__ W,
    const float* __restrict__ bias, float* __restrict__ H, int n) {
    const int lane  = threadIdx.x & 31;
    const int wave  = blockIdx.x * (blockDim.x >> 5) + (threadIdx.x >> 5);
    const int tiles = (n + 15) >> 4;
    if (wave >= tiles) return;                  // wave-uniform: EXEC stays all-1s

    const int mbase = wave << 4;
    const int mrow  = lane & 15;
    const int khalf = (lane >> 4) << 1;         // 0 or 2 within each K-group of 4

    int arow = mbase + mrow;
    if (arow > n - 1) arow = n - 1;             // clamp (tail rows discarded on store)
    const float* xrow = X + (size_t)arow * FI;

    v8f acc[FO / 16];
    v8f zero = {};
#pragma unroll
    for (int t = 0; t < FO / 16; ++t) acc[t] = zero;

#pragma unroll 4
    for (int k0 = 0; k0 < FI; k0 += 4) {
        v2f a = *(const v2f*)(xrow + k0 + khalf);   // two consecutive K values, 8B load
#pragma unroll
        for (int t = 0; t < FO / 16; ++t) {
            const int col = mrow + t * 16;
            v2f b;
            b.x = W[(k0 + khalf + 0) * FO + col];
            b.y = W[(k0 + khalf + 1) * FO + col];
            acc[t] = __builtin_amdgcn_wmma_f32_16x16x4_f32(
                false, a, false, b, (short)0, acc[t], false, false);
        }
    }

    const int rowoff = (lane >> 4) << 3;        // lanes 16..31 hold rows M=8..15
    if (mbase + 16 <= n) {
        // Fast path (all tiles when n % 16 == 0): unguarded, fully coalesced
        // stores — lanes 0..15 of each half-wave cover 64B-contiguous runs.
#pragma unroll
        for (int t = 0; t < FO / 16; ++t) {
            const int col = mrow + t * 16;
            float bv;
            if constexpr (BIAS) bv = bias[col];
#pragma unroll
            for (int r = 0; r < 8; ++r) {
                const int row = mbase + rowoff + r;
                float v = acc[t][r];
                if constexpr (BIAS) v += bv;
                H[(size_t)row * FO + col] = v;
            }
        }
    } else {
#pragma unroll
        for (int t = 0; t < FO / 16; ++t) {
            const int col = mrow + t * 16;
            float bv;
            if constexpr (BIAS) bv = bias[col];
#pragma unroll
            for (int r = 0; r < 8; ++r) {
                const int row = mbase + rowoff + r;
                float v = acc[t][r];
                if constexpr (BIAS) v += bv;
                if (row < n) H[(size_t)row * FO + col] = v;
            }
        }
    }
}

// ---------------------------------------------------------------------------
// Per-node: attention logit halves al_s/al_d = H . a_src / a_dst,
// init segment-max (encoded -inf) and denominator, zero the accumulator O.
// ---------------------------------------------------------------------------
template<int FO>
__global__ __launch_bounds__(256) void node_prep(
    const float* __restrict__ H,
    const float* __restrict__ a_src, const float* __restrict__ a_dst,
    float* __restrict__ alS, float* __restrict__ alD,
    int* __restrict__ mEnc, float* __restrict__ den,
    float* __restrict__ O, int n) {
    const int i = blockIdx.x * blockDim.x + threadIdx.x;
    if (i >= n) return;
    const float* hr = H + (size_t)i * FO;
    float s = 0.0f, d = 0.0f;
#pragma unroll
    for (int f = 0; f < FO; ++f) {
        const float h = hr[f];
        s += h * a_src[f];                      // a_src/a_dst are scalar broadcasts
        d += h * a_dst[f];
    }
    alS[i] = s;
    alD[i] = d;
    mEnc[i] = 0x80000000;                       // encoded "below -inf"
    den[i]  = 0.0f;
    float4* orow = (float4*)(O + (size_t)i * FO);
#pragma unroll
    for (int f = 0; f < FO / 4; ++f) orow[f] = make_float4(0.f, 0.f, 0.f, 0.f);
}

// ---------------------------------------------------------------------------
// Edge pass 1: e = leaky_relu(al_s[src] + al_d[dst]); store e; segment max.
// Edges idx < E come from edge_index; idx >= E are self-loops.
// ---------------------------------------------------------------------------
__global__ __launch_bounds__(256) void edge_logit_max(
    const int* __restrict__ ei, int E, int n,
    const float* __restrict__ alS, const float* __restrict__ alD,
    float* __restrict__ eBuf, int* __restrict__ mEnc) {
    const int idx = blockIdx.x * blockDim.x + threadIdx.x;
    if (idx >= E + n) return;
    int src, dst;
    if (idx < E) { src = ei[idx]; dst = ei[E + idx]; }
    else         { src = dst = idx - E; }
    float e = alS[src] + alD[dst];
    e = e > 0.0f ? e : e * NEG_SLOPE;
    eBuf[idx] = e;
    atomicMax(mEnc + dst, enc_f(e));
}

// ---------------------------------------------------------------------------
// Edge pass 2: ex = exp(e - max[dst]); store ex; segment sum of denominators.
// ---------------------------------------------------------------------------
__global__ __launch_bounds__(256) void edge_exp_sum(
    const int* __restrict__ ei, int E, int n,
    const int* __restrict__ mEnc,
    float* __restrict__ eBuf, float* __restrict__ den) {
    const int idx = blockIdx.x * blockDim.x + threadIdx.x;
    if (idx >= E + n) return;
    int dst = (idx < E) ? ei[E + idx] : (idx - E);
    const float ex = __expf(eBuf[idx] - dec_f(mEnc[dst]));
    eBuf[idx] = ex;
    atomicAdd(den + dst, ex);
}

// ---------------------------------------------------------------------------
// Edge pass 3: alpha = ex/den[dst]; O[dst,:] += alpha * H[src,:].
// float4 gathers of H rows (L2-resident: node arrays << 192MB L2),
// f32 atomic scatter (global_atomic_add_f32).
// ---------------------------------------------------------------------------
template<int FO>
__global__ __launch_bounds__(256) void edge_aggregate(
    const int* __restrict__ ei, int E, int n,
    const float* __restrict__ eBuf, const float* __restrict__ den,
    const float* __restrict__ H, float* __restrict__ O) {
    const int idx = blockIdx.x * blockDim.x + threadIdx.x;
    if (idx >= E + n) return;
    int src, dst;
    if (idx < E) { src = ei[idx]; dst = ei[E + idx]; }
    else         { src = dst = idx - E; }
    const float alpha = eBuf[idx] / den[dst];
    const float4* hr = (const float4*)(H + (size_t)src * FO);
    float* orow = O + (size_t)dst * FO;
#pragma unroll
    for (int q = 0; q < FO / 4; ++q) {
        const float4 h4 = hr[q];
        atomicAdd(orow + 4 * q + 0, alpha * h4.x);
        atomicAdd(orow + 4 * q + 1, alpha * h4.y);
        atomicAdd(orow + 4 * q + 2, alpha * h4.z);
        atomicAdd(orow + 4 * q + 3, alpha * h4.w);
    }
}

// ---------------------------------------------------------------------------
// Per-node: X = relu(O + b)   (in-place safe)
// ---------------------------------------------------------------------------
template<int FO>
__global__ __launch_bounds__(256) void node_finish(
    const float* __restrict__ O, const float* __restrict__ b,
    float* __restrict__ Xn, int n) {
    const int i = blockIdx.x * blockDim.x + threadIdx.x;
    if (i >= n) return;
    const float* orow = O + (size_t)i * FO;
    float* xrow = Xn + (size_t)i * FO;
#pragma unroll
    for (int f = 0; f < FO; ++f) {
        const float v = orow[f] + b[f];
        xrow[f] = v > 0.0f ? v : 0.0f;
    }
}

extern "C" void kernel_launch(void* const* d_in, const int* in_sizes, int n_in,
                              void* d_out, int out_size, void* d_ws, size_t ws_size,
                              hipStream_t stream) {
    const float* x   = (const float*)d_in[0];
    const int*   ei  = (const int*)  d_in[1];
    const float* W1  = (const float*)d_in[2];
    const float* as1 = (const float*)d_in[3];
    const float* ad1 = (const float*)d_in[4];
    const float* b1  = (const float*)d_in[5];
    const float* W2  = (const float*)d_in[6];
    const float* as2 = (const float*)d_in[7];
    const float* ad2 = (const float*)d_in[8];
    const float* b2  = (const float*)d_in[9];
    const float* W3  = (const float*)d_in[10];
    const float* as3 = (const float*)d_in[11];
    const float* ad3 = (const float*)d_in[12];
    const float* b3  = (const float*)d_in[13];
    const float* Wc  = (const float*)d_in[14];
    const float* bc  = (const float*)d_in[15];

    const int n = in_sizes[0] / 128;   // 100000
    const int E = in_sizes[1] / 2;     // 3200000
    const int tot = E + n;

    // Workspace layout (all 4B elements, offsets kept 16B aligned)
    const size_t nr = ((size_t)n + 3) & ~(size_t)3;
    float* ws   = (float*)d_ws;
    float* buf0 = ws;                  // H buffer   [n x 32 max]
    float* buf1 = buf0 + nr * 32;      // X / O ping [n x 32 max]
    float* alS  = buf1 + nr * 32;
    float* alD  = alS + nr;
    int*   mEnc = (int*)(alD + nr);
    float* den  = (float*)(mEnc + nr);
    float* eBuf = den + nr;            // [E + n]

    float* outP = (float*)d_out;       // classifier output [n x 16]
    float* hOut = outP + (size_t)n * 16; // returned h (post-relu layer 3)

    const dim3 blk(256);
    const int nodeBlocks = (n + 255) / 256;
    const int edgeBlocks = (tot + 255) / 256;
    const int tiles      = (n + 15) / 16;
    const int gemmBlocks = (tiles + 7) / 8;    // 8 waves per 256-thread block

    // ---------------- Layer 1: 128 -> 16 ----------------
    wmma_gemm<128, 16, false><<<gemmBlocks, blk, 0, stream>>>(x, W1, nullptr, buf0, n);
    node_prep<16><<<nodeBlocks, blk, 0, stream>>>(buf0, as1, ad1, alS, alD, mEnc, den, buf1, n);
    edge_logit_max<<<edgeBlocks, blk, 0, stream>>>(ei, E, n, alS, alD, eBuf, mEnc);
    edge_exp_sum  <<<edgeBlocks, blk, 0, stream>>>(ei, E, n, mEnc, eBuf, den);
    edge_aggregate<16><<<edgeBlocks, blk, 0, stream>>>(ei, E, n, eBuf, den, buf0, buf1);
    node_finish<16><<<nodeBlocks, blk, 0, stream>>>(buf1, b1, buf1, n);

    // ---------------- Layer 2: 16 -> 32 ----------------
    wmma_gemm<16, 32, false><<<gemmBlocks, blk, 0, stream>>>(buf1, W2, nullptr, buf0, n);
    node_prep<32><<<nodeBlocks, blk, 0, stream>>>(buf0, as2, ad2, alS, alD, mEnc, den, buf1, n);
    edge_logit_max<<<edgeBlocks, blk, 0, stream>>>(ei, E, n, alS, alD, eBuf, mEnc);
    edge_exp_sum  <<<edgeBlocks, blk, 0, stream>>>(ei, E, n, mEnc, eBuf, den);
    edge_aggregate<32><<<edgeBlocks, blk, 0, stream>>>(ei, E, n, eBuf, den, buf0, buf1);
    node_finish<32><<<nodeBlocks, blk, 0, stream>>>(buf1, b2, buf1, n);

    // ---------------- Layer 3: 32 -> 16 ----------------
    wmma_gemm<32, 16, false><<<gemmBlocks, blk, 0, stream>>>(buf1, W3, nullptr, buf0, n);
    node_prep<16><<<nodeBlocks, blk, 0, stream>>>(buf0, as3, ad3, alS, alD, mEnc, den, buf1, n);
    edge_logit_max<<<edgeBlocks, blk, 0, stream>>>(ei, E, n, alS, alD, eBuf, mEnc);
    edge_exp_sum  <<<edgeBlocks, blk, 0, stream>>>(ei, E, n, mEnc, eBuf, den);
    edge_aggregate<16><<<edgeBlocks, blk, 0, stream>>>(ei, E, n, eBuf, den, buf0, buf1);
    node_finish<16><<<nodeBlocks, blk, 0, stream>>>(buf1, b3, hOut, n);   // h output

    // ---------------- Classifier: out = h @ Wc + bc ----------------
    wmma_gemm<16, 16, true><<<gemmBlocks, blk, 0, stream>>>(hOut, Wc, bc, outP, n);
}